// RepFlowLayer_62723702391088
// MI455X (gfx1250) — compile-verified
//
#include <hip/hip_runtime.h>

// ---------------- problem constants ----------------
#define NLOC 1024
#define NALL 2048
#define N_DIM 128
#define E_DIM 16
#define A_DIM 64
#define NNEI 120
#define A_SEL 20

// workspace layout (float offsets)
constexpr int OFF_PRE   = 0;               // 1024*352
constexpr int OFF_SYMIN = 360448;          // 1024*576
constexpr int OFF_EACC  = 950272;          // 1024*128
constexpr int OFF_BPN   = 1081344;         // packed (128 x 352)  -> 45056
constexpr int OFF_BPE   = 1126400;         // packed (144 x 144)  -> 20736
constexpr int OFF_BPS   = 1147136;         // packed (576 x 128)  -> 73728
constexpr int OFF_BPA   = 1220864;         // packed (64 x 80)    -> 5120
constexpr int OFF_WJ    = 1225984;         // raw 16x80
constexpr int OFF_WK    = 1227264;         // raw 16x80

typedef float v2f __attribute__((ext_vector_type(2)));
typedef float v8f __attribute__((ext_vector_type(8)));

__device__ __forceinline__ float silu_f(float x) { return x / (1.0f + __expf(-x)); }

__device__ __forceinline__ v8f wmma_f32(v2f a, v2f b, v8f c) {
  // V_WMMA_F32_16X16X4_F32 : D = A(16x4) * B(4x16) + C(16x16), full fp32
  return __builtin_amdgcn_wmma_f32_16x16x4_f32(false, a, false, b, (short)0, c,
                                               false, false);
}

// B-fragment packed index: element B[k][n] of a KxN matrix lives at
// (((k>>2)*2 + ((k>>1)&1))*N + n)*2 + (k&1)
// lane l of a wave loads float2 at ((kt*2 + (l>>4))*N + n0 + (l&15))*2,
// i.e. per-lane base (hf*N + n0 + l16)*2 plus kt*4N floats (imm offsets).

// ---------------- weight packing ----------------
__global__ void k_pack(const float* __restrict__ wns,   // w_node_self 128x128
                       const float* __restrict__ wne,   // w_node_edge 272x128
                       const float* __restrict__ wes,   // w_edge_self 272x16
                       const float* __restrict__ wsym,  // w_node_sym 576x128
                       const float* __restrict__ wea1,  // w_ea1 224x16
                       const float* __restrict__ was,   // w_angle_self 224x64
                       float* __restrict__ ws) {
  int t = blockIdx.x * blockDim.x + threadIdx.x;
  if (t < 45056) {  // BpN: K=128, N=352
    int e = t, j = e & 1, tmp = e >> 1;
    int n = tmp % 352, q = tmp / 352;
    int k = 4 * (q >> 1) + 2 * (q & 1) + j;
    float v;
    if (n < 128)      v = wns[k * 128 + n];
    else if (n < 256) v = wne[k * 128 + (n - 128)];
    else if (n < 272) v = wes[k * 16 + (n - 256)];
    else if (n < 288) v = wea1[(64 + k) * 16 + (n - 272)];
    else              v = was[(64 + k) * 64 + (n - 288)];
    ws[OFF_BPN + e] = v;
  } else if (t < 65792) {  // BpE: K=144, N=144 (rows 128..271 of edge weights)
    int e = t - 45056, j = e & 1, tmp = e >> 1;
    int n = tmp % 144, q = tmp / 144;
    int k = 4 * (q >> 1) + 2 * (q & 1) + j;
    float v = (n < 128) ? wne[(128 + k) * 128 + n] : wes[(128 + k) * 16 + (n - 128)];
    ws[OFF_BPE + e] = v;
  } else if (t < 139520) {  // BpS: K=576, N=128
    int e = t - 65792, j = e & 1, tmp = e >> 1;
    int n = tmp % 128, q = tmp / 128;
    int k = 4 * (q >> 1) + 2 * (q & 1) + j;
    ws[OFF_BPS + e] = wsym[k * 128 + n];
  } else if (t < 144640) {  // BpA: K=64, N=80 (rows 0..63 of angle weights)
    int e = t - 139520, j = e & 1, tmp = e >> 1;
    int n = tmp % 80, q = tmp / 80;
    int k = 4 * (q >> 1) + 2 * (q & 1) + j;
    float v = (n < 16) ? wea1[k * 16 + n] : was[k * 64 + (n - 16)];
    ws[OFF_BPA + e] = v;
  } else if (t < 145920) {  // WjRaw: e_ik rows 192..207
    int e = t - 144640, d = e / 80, c = e - d * 80;
    ws[OFF_WJ + e] = (c < 16) ? wea1[(192 + d) * 16 + c] : was[(192 + d) * 64 + (c - 16)];
  } else if (t < 147200) {  // WkRaw: e_ij rows 208..223
    int e = t - 145920, d = e / 80, c = e - d * 80;
    ws[OFF_WK + e] = (c < 16) ? wea1[(208 + d) * 16 + c] : was[(208 + d) * 64 + (c - 16)];
  }
}

// ---------------- node pre-GEMM: (1024,128)@(128,352) -> pre ----------------
__global__ __launch_bounds__(256) void k_pre(const float* __restrict__ node,
                                             const float* __restrict__ bpn,
                                             float* __restrict__ pre) {
  int i0 = blockIdx.x * 16;
  int lane = threadIdx.x & 31, wv = threadIdx.x >> 5;
  int hf = lane >> 4, l16 = lane & 15;
  // cache A fragments (K=128 -> 32 steps) in registers; reused for all N-tiles
  const float* ap = node + (i0 + l16) * 128 + 2 * hf;
  v2f af[32];
#pragma unroll
  for (int kt = 0; kt < 32; ++kt) af[kt] = *(const v2f*)(ap + kt * 4);
  for (int nt = wv; nt < 22; nt += 8) {
    int n0 = nt * 16;
    const float* bwp = bpn + (hf * 352 + n0 + l16) * 2;
    v8f acc = {0.f, 0.f, 0.f, 0.f, 0.f, 0.f, 0.f, 0.f};
#pragma unroll
    for (int kt = 0; kt < 32; ++kt)
      acc = wmma_f32(af[kt], *(const v2f*)(bwp + kt * 1408), acc);
    for (int j = 0; j < 8; ++j) {
      int m = i0 + j + 8 * hf;
      pre[m * 352 + n0 + l16] = acc[j];
    }
  }
}

// ---------------- per-atom edge kernel ----------------
// A (LDS, 128x148 padded): rows r<120 = [masked nei_node(128) | edge_ebd(16)]
// GEMM vs packed (144x144). Also computes sym 576-vector and node_edge acc.
__global__ __launch_bounds__(256) void k_edge(
    const float* __restrict__ node_ext, const float* __restrict__ edge,
    const float* __restrict__ h2, const float* __restrict__ sw,
    const int* __restrict__ nlist, const unsigned char* __restrict__ nmask,
    const float* __restrict__ bpe, const float* __restrict__ pre,
    const float* __restrict__ b_ne, const float* __restrict__ b_es,
    const float* __restrict__ e_res, float* __restrict__ out_edge,
    float* __restrict__ symin, float* __restrict__ eacc) {
  const int i = blockIdx.x;
  const int tid = threadIdx.x;
  __shared__ float As[128 * 148];
  __shared__ float s_h2[360];
  __shared__ float s_sw[128];
  __shared__ float s_wgt[120];
  __shared__ int s_nl[120];
  __shared__ unsigned char s_m[120];
  __shared__ float s_hg[3 * 144];
  __shared__ float s_acc[128];

  if (tid < 128) s_acc[tid] = 0.f;
  if (tid < 120) {
    float swv = sw[i * NNEI + tid];
    unsigned char m = nmask[i * NNEI + tid];
    s_sw[tid] = swv;
    s_wgt[tid] = m ? swv : 0.f;
    s_nl[tid] = nlist[i * NNEI + tid];
    s_m[tid] = m;
  } else if (tid < 128) {
    s_sw[tid] = 0.f;
  }
  for (int t = tid; t < 360; t += 256) s_h2[t] = h2[i * 360 + t];
  __syncthreads();

  for (int idx = tid; idx < 128 * 144; idx += 256) {
    int r = idx / 144, c = idx - r * 144;
    float v = 0.f;
    if (r < 120) {
      if (c < 128) v = s_m[r] ? node_ext[s_nl[r] * 128 + c] : 0.f;
      else         v = edge[(i * NNEI + r) * 16 + (c - 128)];
    }
    As[r * 148 + c] = v;
  }
  __syncthreads();

  // hg[i3][d] = sum_r h2[r][i3]*wgt[r]*A[r][d] / 120, d in [0,144)
  if (tid < 144) {
    for (int i3 = 0; i3 < 3; ++i3) {
      float s = 0.f;
      for (int r = 0; r < 120; ++r)
        s += s_h2[r * 3 + i3] * s_wgt[r] * As[r * 148 + tid];
      s_hg[i3 * 144 + tid] = s * (1.f / 120.f);
    }
  }
  __syncthreads();

  // grrg -> node_sym_in (576)
  for (int o = tid; o < 576; o += 256) {
    int d, a;
    if (o < 512) { d = o >> 2; a = o & 3; }
    else { int o2 = o - 512; d = 128 + (o2 >> 2); a = 128 + (o2 & 3); }
    float s = (s_hg[d] * s_hg[a] + s_hg[144 + d] * s_hg[144 + a] +
               s_hg[288 + d] * s_hg[288 + a]) * (1.f / 3.f);
    symin[i * 576 + o] = s;
  }

  // WMMA: wave w owns M-tile w; A fragments cached in regs, 9 N-tiles
  int lane = tid & 31, wv = tid >> 5;
  int hf = lane >> 4, l16 = lane & 15;
  const int mt = wv;
  const float* awp = &As[(mt * 16 + l16) * 148 + 2 * hf];
  v2f af[36];
#pragma unroll
  for (int kt = 0; kt < 36; ++kt) af[kt] = *(const v2f*)(awp + kt * 4);

  for (int nt = 0; nt < 9; ++nt) {
    int n0 = nt * 16;
    const float* bwp = bpe + (hf * 144 + n0 + l16) * 2;
    v8f acc = {0.f, 0.f, 0.f, 0.f, 0.f, 0.f, 0.f, 0.f};
#pragma unroll
    for (int kt = 0; kt < 36; ++kt)
      acc = wmma_f32(af[kt], *(const v2f*)(bwp + kt * 576), acc);
    int n = n0 + l16;
    float prev = pre[i * 352 + 128 + n];  // node-row contribution
    if (nt < 8) {  // node_edge_update columns
      float bias = b_ne[n];
      float ssum = 0.f;
      for (int j = 0; j < 8; ++j) {
        int r = mt * 16 + j + 8 * hf;
        if (r < 120) ssum += silu_f(acc[j] + prev + bias) * s_sw[r];
      }
      atomicAdd(&s_acc[n], ssum);
    } else {  // edge_self_update columns
      int ce = n - 128;
      float bias = b_es[ce];
      float er = e_res[ce];  // e_res[0][ce]
      for (int j = 0; j < 8; ++j) {
        int r = mt * 16 + j + 8 * hf;
        if (r < 120) {
          float u = silu_f(acc[j] + prev + bias);
          int gi = (i * NNEI + r) * 16 + ce;
          out_edge[gi] = edge[gi] + er * u;
        }
      }
    }
  }
  __syncthreads();
  if (tid < 128) eacc[i * 128 + tid] = s_acc[tid];
}

// ---------------- sym GEMM + node_new epilogue ----------------
__global__ __launch_bounds__(256) void k_sym(
    const float* __restrict__ symin, const float* __restrict__ bps,
    const float* __restrict__ pre, const float* __restrict__ node_ext,
    const float* __restrict__ b_ns, const float* __restrict__ b_nsym,
    const float* __restrict__ n_res, const float* __restrict__ eacc,
    float* __restrict__ out_node) {
  int i0 = blockIdx.x * 16;
  int tid = threadIdx.x;
  int lane = tid & 31, nt = tid >> 5;  // 8 waves = 8 N-tiles
  int hf = lane >> 4, l16 = lane & 15;
  int n = nt * 16 + l16;
  int arow = i0 + l16;
  const float* ap = symin + arow * 576 + 2 * hf;
  const float* bwp = bps + (hf * 128 + n) * 2;
  v8f acc = {0.f, 0.f, 0.f, 0.f, 0.f, 0.f, 0.f, 0.f};
#pragma unroll 16
  for (int kt = 0; kt < 144; ++kt)
    acc = wmma_f32(*(const v2f*)(ap + kt * 4), *(const v2f*)(bwp + kt * 512),
                   acc);
  float bs = b_ns[n], bm = b_nsym[n];
  float r0 = n_res[n], r1 = n_res[128 + n], r2 = n_res[256 + n];
  for (int j = 0; j < 8; ++j) {
    int m = i0 + j + 8 * hf;
    float v = node_ext[m * 128 + n] + r0 * silu_f(pre[m * 352 + n] + bs) +
              r1 * silu_f(acc[j] + bm) +
              r2 * (eacc[m * 128 + n] * (1.f / 120.f));
    out_node[m * 128 + n] = v;
  }
}

// ---------------- per-atom angle kernel ----------------
__global__ __launch_bounds__(256) void k_angle(
    const float* __restrict__ angle, const float* __restrict__ edge,
    const float* __restrict__ a_sw, const unsigned char* __restrict__ amask,
    const float* __restrict__ pre, const float* __restrict__ bpa,
    const float* __restrict__ wjr, const float* __restrict__ wkr,
    const float* __restrict__ b_ea1, const float* __restrict__ b_as,
    const float* __restrict__ w_ea2, const float* __restrict__ b_ea2,
    const float* __restrict__ e_res, const float* __restrict__ a_res,
    float* __restrict__ out_edge, float* __restrict__ out_angle) {
  const int i = blockIdx.x;
  const int tid = threadIdx.x;
  __shared__ float Aa[400 * 66];
  __shared__ float s_ej[20 * 80];
  __shared__ float s_ek[20 * 80];
  __shared__ float s_red[20 * 16];
  __shared__ float s_e[20 * 16];
  __shared__ float s_asw[20];
  __shared__ float s_w2[256];
  __shared__ unsigned char s_am[20];

  if (tid < 20) {
    s_asw[tid] = a_sw[i * 20 + tid];
    s_am[tid] = amask[i * 20 + tid];
  }
  if (tid < 256) s_w2[tid] = w_ea2[tid];
  for (int t = tid; t < 320; t += 256) {
    int j = t >> 4, d = t & 15;
    s_red[t] = 0.f;
    unsigned char m = amask[i * 20 + j];
    s_e[t] = m ? edge[(i * NNEI + j) * 16 + d] : 0.f;
  }
  for (int idx = tid; idx < 400 * 64; idx += 256) {
    int r = idx >> 6, c = idx & 63;
    Aa[r * 66 + c] = angle[i * 25600 + idx];
  }
  __syncthreads();

  // per-j (e_ik) and per-k (e_ij) 16-dim corrections; fold node-term + bias into ej
  for (int o = tid; o < 1600; o += 256) {
    int j = o / 80, c = o - j * 80;
    float sej = pre[i * 352 + 272 + c] + (c < 16 ? b_ea1[c] : b_as[c - 16]);
    float sek = 0.f;
#pragma unroll
    for (int d = 0; d < 16; ++d) {
      float ev = s_e[j * 16 + d];
      sej += ev * wjr[d * 80 + c];
      sek += ev * wkr[d * 80 + c];
    }
    s_ej[o] = sej;
    s_ek[o] = sek;
  }
  __syncthreads();

  int lane = tid & 31, wv = tid >> 5;
  int hf = lane >> 4, l16 = lane & 15;
  // wave w owns M-tiles w, w+8, w+16 (and 24 for wave 0); A fragments cached across 5 N-tiles
  for (int mt = wv; mt < 25; mt += 8) {
    const float* awp = &Aa[(mt * 16 + l16) * 66 + 2 * hf];
    v2f af[16];
#pragma unroll
    for (int kt = 0; kt < 16; ++kt) af[kt] = *(const v2f*)(awp + kt * 4);
    for (int nt = 0; nt < 5; ++nt) {
      int n0 = nt * 16;
      const float* bwp = bpa + (hf * 80 + n0 + l16) * 2;
      v8f acc = {0.f, 0.f, 0.f, 0.f, 0.f, 0.f, 0.f, 0.f};
#pragma unroll
      for (int kt = 0; kt < 16; ++kt)
        acc = wmma_f32(af[kt], *(const v2f*)(bwp + kt * 320), acc);
      int c = n0 + l16;
      for (int j = 0; j < 8; ++j) {
        int rho = mt * 16 + j + 8 * hf;
        int ja = rho / 20, ka = rho - ja * 20;
        float val = acc[j] + s_ej[ja * 80 + c] + s_ek[ka * 80 + c];
        if (c < 16) {  // ea path
          float ea = silu_f(val);
          atomicAdd(&s_red[ja * 16 + c], s_asw[ja] * s_asw[ka] * ea);
        } else {  // angle_self path
          int cc = c - 16;
          out_angle[(i * 400 + rho) * 64 + cc] =
              Aa[rho * 66 + cc] + a_res[cc] * silu_f(val);
        }
      }
    }
  }
  __syncthreads();

  // edge_angle_update: padded @ w_ea2 (+bias, silu), RMW into edge output
  if (tid < 120) {
    int nn = tid;
    bool use_red = (nn < 20) && s_am[nn];
    const float sc = 0.22360679774997896f;  // 1/sqrt(20)
    float p[16];
#pragma unroll
    for (int d = 0; d < 16; ++d)
      p[d] = use_red ? s_red[nn * 16 + d] * sc
                     : edge[(i * NNEI + nn) * 16 + d];
#pragma unroll
    for (int c = 0; c < 16; ++c) {
      float s = b_ea2[c];
#pragma unroll
      for (int d = 0; d < 16; ++d) s += p[d] * s_w2[d * 16 + c];
      float u = silu_f(s);
      int gi = (i * NNEI + nn) * 16 + c;
      out_edge[gi] = out_edge[gi] + e_res[16 + c] * u;  // e_res[1][c]
    }
  }
}

// ---------------- launch ----------------
extern "C" void kernel_launch(void* const* d_in, const int* in_sizes, int n_in,
                              void* d_out, int out_size, void* d_ws,
                              size_t ws_size, hipStream_t stream) {
  const float* node_ext = (const float*)d_in[0];
  const float* edge     = (const float*)d_in[1];
  const float* h2       = (const float*)d_in[2];
  const float* angle    = (const float*)d_in[3];
  const float* sw       = (const float*)d_in[4];
  const float* a_sw     = (const float*)d_in[5];
  const int* nlist      = (const int*)d_in[6];
  const unsigned char* nmask = (const unsigned char*)d_in[7];
  const unsigned char* amask = (const unsigned char*)d_in[8];
  const float* w_ns  = (const float*)d_in[9];
  const float* b_ns  = (const float*)d_in[10];
  const float* w_sym = (const float*)d_in[11];
  const float* b_sym = (const float*)d_in[12];
  const float* w_ne  = (const float*)d_in[13];
  const float* b_ne  = (const float*)d_in[14];
  const float* w_es  = (const float*)d_in[15];
  const float* b_es  = (const float*)d_in[16];
  const float* w_ea1 = (const float*)d_in[17];
  const float* b_ea1 = (const float*)d_in[18];
  const float* w_ea2 = (const float*)d_in[19];
  const float* b_ea2 = (const float*)d_in[20];
  const float* w_as  = (const float*)d_in[21];
  const float* b_as  = (const float*)d_in[22];
  const float* n_res = (const float*)d_in[23];
  const float* e_res = (const float*)d_in[24];
  const float* a_res = (const float*)d_in[25];

  float* ws = (float*)d_ws;
  float* pre   = ws + OFF_PRE;
  float* symin = ws + OFF_SYMIN;
  float* eacc  = ws + OFF_EACC;
  float* bpn   = ws + OFF_BPN;
  float* bpe   = ws + OFF_BPE;
  float* bps   = ws + OFF_BPS;
  float* bpa   = ws + OFF_BPA;
  float* wjr   = ws + OFF_WJ;
  float* wkr   = ws + OFF_WK;

  float* out_node  = (float*)d_out;
  float* out_edge  = out_node + NLOC * N_DIM;
  float* out_angle = out_edge + NLOC * NNEI * E_DIM;

  k_pack<<<575, 256, 0, stream>>>(w_ns, w_ne, w_es, w_sym, w_ea1, w_as, ws);
  k_pre<<<64, 256, 0, stream>>>(node_ext, bpn, pre);
  k_edge<<<1024, 256, 0, stream>>>(node_ext, edge, h2, sw, nlist, nmask, bpe,
                                   pre, b_ne, b_es, e_res, out_edge, symin,
                                   eacc);
  k_sym<<<64, 256, 0, stream>>>(symin, bps, pre, node_ext, b_ns, b_sym, n_res,
                                eacc, out_node);
  k_angle<<<1024, 256, 0, stream>>>(angle, edge, a_sw, amask, pre, bpa, wjr,
                                    wkr, b_ea1, b_as, w_ea2, b_ea2, e_res,
                                    a_res, out_edge, out_angle);
}